// GAMMA_GNN_14370960572512
// MI455X (gfx1250) — compile-verified
//
#include <hip/hip_runtime.h>

// GAMMA-GNN forward for MI455X (gfx1250, wave32).
// degree -> [scatter-mean x2 -> fused 3-power WMMA GEMM (LDS-staged A,
// transposed-W B) -> LN] x2 -> WMMA head GEMM -> log-softmax.
// Matrix math: V_WMMA_F32_16X16X4_F32 (full fp32). Scatter: global_atomic_add_f32.
// Working set is L2-resident (192MB); scatter/atomics dominate, GEMM is cheap.

#define HIDC 128
#define KDIM 128
#define LPITCH 132  // LDS row pitch in floats: stride%64==4 -> conflict-free b64 reads

typedef float v2f __attribute__((ext_vector_type(2)));
typedef float v8f __attribute__((ext_vector_type(8)));

static __device__ __forceinline__ void atomic_add_f32(float* p, float v) {
  __hip_atomic_fetch_add(p, v, __ATOMIC_RELAXED, __HIP_MEMORY_SCOPE_AGENT);
}

__global__ __launch_bounds__(256) void k_zero(float* __restrict__ p, long n) {
  long i = (long)blockIdx.x * blockDim.x + threadIdx.x;
  if (i < n) p[i] = 0.0f;
}

__global__ __launch_bounds__(256) void k_degree(const int* __restrict__ dst,
                                                float* __restrict__ deg, int E) {
  int e = blockIdx.x * blockDim.x + threadIdx.x;
  if (e < E) atomic_add_f32(&deg[dst[e]], 1.0f);
}

__global__ __launch_bounds__(256) void k_invdeg(float* __restrict__ deg, int N) {
  int i = blockIdx.x * blockDim.x + threadIdx.x;
  if (i < N) deg[i] = 1.0f / fmaxf(deg[i], 1.0f);
}

// One wave per edge: coalesced 512B row gather, 4 f32 atomics per lane.
__global__ __launch_bounds__(256) void k_scatter_add(const int* __restrict__ src,
                                                     const int* __restrict__ dst,
                                                     const float* __restrict__ h,
                                                     float* __restrict__ out, int E) {
  int lane = threadIdx.x & 31;
  int e = blockIdx.x * 8 + (threadIdx.x >> 5);
  if (e >= E) return;
  int s = src[e];
  int d = dst[e];
  float4 v = reinterpret_cast<const float4*>(h + (size_t)s * HIDC)[lane];
  float* o = out + (size_t)d * HIDC + lane * 4;
  atomic_add_f32(o + 0, v.x);
  atomic_add_f32(o + 1, v.y);
  atomic_add_f32(o + 2, v.z);
  atomic_add_f32(o + 3, v.w);
}

__global__ __launch_bounds__(256) void k_scale_rows(float* __restrict__ buf,
                                                    const float* __restrict__ inv,
                                                    long total) {
  long i = (long)blockIdx.x * blockDim.x + threadIdx.x;
  if (i < total) buf[i] *= inv[i >> 7];  // 128 channels per row
}

// Wt[p][n][k] = W[p][k][n]  (makes B-operand fetches contiguous float2)
__global__ __launch_bounds__(256) void k_transpose_w(const float* __restrict__ W,
                                                     float* __restrict__ Wt,
                                                     int K, int NOUT, int P) {
  long total = (long)P * K * NOUT;
  long i = (long)blockIdx.x * blockDim.x + threadIdx.x;
  if (i >= total) return;
  int p = (int)(i / (K * NOUT));
  int rem = (int)(i % (K * NOUT));
  int k = rem / NOUT;
  int n = rem % NOUT;
  Wt[(size_t)p * K * NOUT + (size_t)n * K + k] = W[i];
}

// Fused 3-power GEMM: out = X0@W0 + X1@W1 + X2@W2 + bias (all 128x128).
// Block = 16-row stripe; 8 waves each own a 16-col tile. A tiles staged in LDS.
// A operand (16x4 f32): lanes 0-15 -> K {k,k+1}; lanes 16-31 -> K {k+2,k+3}.
// B operand mirrors the K split; C/D: VGPR i -> M=i(+8 upper half), N=lane&15.
__global__ __launch_bounds__(256) void k_wmma_gemm3(
    const float* __restrict__ X0, const float* __restrict__ X1,
    const float* __restrict__ X2, const float* __restrict__ Wt,
    const float* __restrict__ bias, float* __restrict__ out) {
  __shared__ float lds[3 * 16 * LPITCH];
  const int t = threadIdx.x;
  const int lane = t & 31;
  const int mbase = blockIdx.x * 16;

  // Cooperative stage: three 16x128 A tiles -> LDS (8 floats per thread per tile).
  {
    const int r = t >> 4;          // 0..15
    const int c = (t & 15) * 8;    // 0,8,...,120
    const float4* g0 = reinterpret_cast<const float4*>(X0 + (size_t)(mbase + r) * KDIM + c);
    const float4* g1 = reinterpret_cast<const float4*>(X1 + (size_t)(mbase + r) * KDIM + c);
    const float4* g2 = reinterpret_cast<const float4*>(X2 + (size_t)(mbase + r) * KDIM + c);
    float4* l0 = reinterpret_cast<float4*>(lds + 0 * 16 * LPITCH + r * LPITCH + c);
    float4* l1 = reinterpret_cast<float4*>(lds + 1 * 16 * LPITCH + r * LPITCH + c);
    float4* l2 = reinterpret_cast<float4*>(lds + 2 * 16 * LPITCH + r * LPITCH + c);
    l0[0] = g0[0]; l0[1] = g0[1];
    l1[0] = g1[0]; l1[1] = g1[1];
    l2[0] = g2[0]; l2[1] = g2[1];
  }
  __syncthreads();

  const int nbase = (t >> 5) * 16;
  const int m = lane & 15;
  const int ncol = nbase + m;
  const int koff = (lane < 16) ? 0 : 2;

  v8f acc = {};
#pragma unroll
  for (int p = 0; p < 3; ++p) {
    const float* la = lds + p * 16 * LPITCH + m * LPITCH + koff;
    const float* wb = Wt + (size_t)p * (KDIM * HIDC) + (size_t)ncol * KDIM + koff;
#pragma unroll 8
    for (int k = 0; k < KDIM; k += 4) {
      v2f a, b;
      float2 av = *reinterpret_cast<const float2*>(la + k);
      float2 bv = *reinterpret_cast<const float2*>(wb + k);
      a[0] = av.x; a[1] = av.y;
      b[0] = bv.x; b[1] = bv.y;
      acc = __builtin_amdgcn_wmma_f32_16x16x4_f32(
          false, a, false, b, (short)0, acc, false, false);
    }
  }

  const float bv = bias[ncol];
  const int mo = (lane < 16) ? 0 : 8;
  float* orow = out + (size_t)mbase * HIDC + ncol;
#pragma unroll
  for (int i = 0; i < 8; ++i) orow[(size_t)(i + mo) * HIDC] = acc[i] + bv;
}

// Head GEMM: logits = H @ linW + bias (128 -> 64). 4 waves per 16-row stripe.
__global__ __launch_bounds__(128) void k_wmma_head(
    const float* __restrict__ X, const float* __restrict__ Wt,
    const float* __restrict__ bias, float* __restrict__ out, int NOUT) {
  __shared__ float lds[16 * LPITCH];
  const int t = threadIdx.x;
  const int lane = t & 31;
  const int mbase = blockIdx.x * 16;

  {
    const int r = t >> 3;         // 0..15
    const int c = (t & 7) * 16;   // 0,16,...,112
    const float4* g = reinterpret_cast<const float4*>(X + (size_t)(mbase + r) * KDIM + c);
    float4* l = reinterpret_cast<float4*>(lds + r * LPITCH + c);
    l[0] = g[0]; l[1] = g[1]; l[2] = g[2]; l[3] = g[3];
  }
  __syncthreads();

  const int nbase = (t >> 5) * 16;
  const int m = lane & 15;
  const int ncol = nbase + m;
  const int koff = (lane < 16) ? 0 : 2;

  v8f acc = {};
  const float* la = lds + m * LPITCH + koff;
  const float* wb = Wt + (size_t)ncol * KDIM + koff;
#pragma unroll 8
  for (int k = 0; k < KDIM; k += 4) {
    v2f a, b;
    float2 av = *reinterpret_cast<const float2*>(la + k);
    float2 bv = *reinterpret_cast<const float2*>(wb + k);
    a[0] = av.x; a[1] = av.y;
    b[0] = bv.x; b[1] = bv.y;
    acc = __builtin_amdgcn_wmma_f32_16x16x4_f32(
        false, a, false, b, (short)0, acc, false, false);
  }

  const float bv = bias[ncol];
  const int mo = (lane < 16) ? 0 : 8;
  float* orow = out + (size_t)mbase * NOUT + ncol;
#pragma unroll
  for (int i = 0; i < 8; ++i) orow[(size_t)(i + mo) * NOUT] = acc[i] + bv;
}

// Wave-per-row LayerNorm over 128 channels (float4 per lane + shfl_xor tree).
__global__ __launch_bounds__(256) void k_layernorm(float* __restrict__ buf,
                                                   const float* __restrict__ gw,
                                                   const float* __restrict__ gb,
                                                   int Nrows) {
  int lane = threadIdx.x & 31;
  int row = blockIdx.x * 8 + (threadIdx.x >> 5);
  if (row >= Nrows) return;
  float4* rp = reinterpret_cast<float4*>(buf) + (size_t)row * (HIDC / 4);
  float4 v = rp[lane];
  float s = v.x + v.y + v.z + v.w;
  float q = v.x * v.x + v.y * v.y + v.z * v.z + v.w * v.w;
#pragma unroll
  for (int msk = 16; msk >= 1; msk >>= 1) {
    s += __shfl_xor(s, msk, 32);
    q += __shfl_xor(q, msk, 32);
  }
  float mu = s * (1.0f / HIDC);
  float var = q * (1.0f / HIDC) - mu * mu;
  float rs = rsqrtf(var + 1e-5f);
  float4 g4 = reinterpret_cast<const float4*>(gw)[lane];
  float4 b4 = reinterpret_cast<const float4*>(gb)[lane];
  v.x = (v.x - mu) * rs * g4.x + b4.x;
  v.y = (v.y - mu) * rs * g4.y + b4.y;
  v.z = (v.z - mu) * rs * g4.z + b4.z;
  v.w = (v.w - mu) * rs * g4.w + b4.w;
  rp[lane] = v;
}

// Wave-per-row log-softmax over 64 logits (2 per lane), in place.
__global__ __launch_bounds__(256) void k_logsoftmax(float* __restrict__ out,
                                                    int Nrows, int C) {
  int lane = threadIdx.x & 31;
  int row = blockIdx.x * 8 + (threadIdx.x >> 5);
  if (row >= Nrows) return;
  float* r = out + (size_t)row * C;
  float a = r[lane];
  float b = r[lane + 32];
  float mx = fmaxf(a, b);
#pragma unroll
  for (int m = 16; m >= 1; m >>= 1) mx = fmaxf(mx, __shfl_xor(mx, m, 32));
  float e = expf(a - mx) + expf(b - mx);
#pragma unroll
  for (int m = 16; m >= 1; m >>= 1) e += __shfl_xor(e, m, 32);
  float lse = mx + logf(e);
  r[lane] = a - lse;
  r[lane + 32] = b - lse;
}

extern "C" void kernel_launch(void* const* d_in, const int* in_sizes, int n_in,
                              void* d_out, int out_size, void* d_ws, size_t ws_size,
                              hipStream_t stream) {
  (void)n_in; (void)out_size; (void)ws_size;
  const float* x    = (const float*)d_in[0];
  const int*   ei   = (const int*)d_in[1];
  const float* W1   = (const float*)d_in[2];
  const float* b1   = (const float*)d_in[3];
  const float* W2   = (const float*)d_in[4];
  const float* b2   = (const float*)d_in[5];
  const float* ln1g = (const float*)d_in[6];
  const float* ln1b = (const float*)d_in[7];
  const float* ln2g = (const float*)d_in[8];
  const float* ln2b = (const float*)d_in[9];
  const float* linW = (const float*)d_in[10];
  const float* linb = (const float*)d_in[11];
  float* outp = (float*)d_out;

  const int N = in_sizes[0] / HIDC;   // 50000 (16 | N)
  const int E = in_sizes[1] / 2;      // 600000
  const int OUTC = in_sizes[11];      // 64
  const int* src = ei;
  const int* dstp = ei + E;

  const long NH = (long)N * HIDC;
  const long W3 = 3L * KDIM * HIDC;       // 49152
  const long WH = (long)KDIM * OUTC;      // 8192
  float* ws = (float*)d_ws;
  long pad = ((long)N + 255) & ~255L;
  float* inv_deg = ws;           // N
  float* agg1 = ws + pad;        // N*128
  float* agg2 = agg1 + NH;       // N*128
  float* bufA = agg2 + NH;       // N*128 (layer-1 activations)
  float* bufB = bufA + NH;       // N*128 (layer-2 activations)
  float* Wt1  = bufB + NH;       // 3*128*128 transposed
  float* Wt2  = Wt1 + W3;        // 3*128*128 transposed
  float* WtH  = Wt2 + W3;        // 64*128 transposed

  dim3 b256(256);
  int nhBlocks = (int)((NH + 255) / 256);
  int nBlocks = (N + 255) / 256;
  int eBlocks = (E + 255) / 256;
  int eWaveBlocks = (E + 7) / 8;   // wave per edge
  int rowBlocks = (N + 7) / 8;     // wave per row
  int mTiles = N / 16;             // 3125

  // weight transposes (tiny, once per launch)
  k_transpose_w<<<(int)((W3 + 255) / 256), b256, 0, stream>>>(W1, Wt1, KDIM, HIDC, 3);
  k_transpose_w<<<(int)((W3 + 255) / 256), b256, 0, stream>>>(W2, Wt2, KDIM, HIDC, 3);
  k_transpose_w<<<(int)((WH + 255) / 256), b256, 0, stream>>>(linW, WtH, KDIM, OUTC, 1);

  // degrees -> inv_deg
  k_zero<<<nBlocks, b256, 0, stream>>>(inv_deg, N);
  k_degree<<<eBlocks, b256, 0, stream>>>(dstp, inv_deg, E);
  k_invdeg<<<nBlocks, b256, 0, stream>>>(inv_deg, N);

  // ---- layer 1 ----
  k_zero<<<nhBlocks, b256, 0, stream>>>(agg1, NH);
  k_scatter_add<<<eWaveBlocks, b256, 0, stream>>>(src, dstp, x, agg1, E);
  k_scale_rows<<<nhBlocks, b256, 0, stream>>>(agg1, inv_deg, NH);
  k_zero<<<nhBlocks, b256, 0, stream>>>(agg2, NH);
  k_scatter_add<<<eWaveBlocks, b256, 0, stream>>>(src, dstp, agg1, agg2, E);
  k_scale_rows<<<nhBlocks, b256, 0, stream>>>(agg2, inv_deg, NH);
  k_wmma_gemm3<<<mTiles, b256, 0, stream>>>(x, agg1, agg2, Wt1, b1, bufA);
  k_layernorm<<<rowBlocks, b256, 0, stream>>>(bufA, ln1g, ln1b, N);

  // ---- layer 2 ----
  k_zero<<<nhBlocks, b256, 0, stream>>>(agg1, NH);
  k_scatter_add<<<eWaveBlocks, b256, 0, stream>>>(src, dstp, bufA, agg1, E);
  k_scale_rows<<<nhBlocks, b256, 0, stream>>>(agg1, inv_deg, NH);
  k_zero<<<nhBlocks, b256, 0, stream>>>(agg2, NH);
  k_scatter_add<<<eWaveBlocks, b256, 0, stream>>>(src, dstp, agg1, agg2, E);
  k_scale_rows<<<nhBlocks, b256, 0, stream>>>(agg2, inv_deg, NH);
  k_wmma_gemm3<<<mTiles, b256, 0, stream>>>(bufA, agg1, agg2, Wt2, b2, bufB);
  k_layernorm<<<rowBlocks, b256, 0, stream>>>(bufB, ln2g, ln2b, N);

  // ---- head: 128 -> 64 linear + log-softmax (in place on d_out) ----
  k_wmma_head<<<mTiles, dim3(128), 0, stream>>>(bufB, WtH, linb, outp, OUTC);
  k_logsoftmax<<<rowBlocks, b256, 0, stream>>>(outp, N, OUTC);
}